// GNNBlock_88390426952001
// MI455X (gfx1250) — compile-verified
//
#include <hip/hip_runtime.h>

// GNN block: GCNConv (f32 WMMA GEMM + normalized scatter-add) + BatchNorm + ReLU + residual.
// MI455X / gfx1250, wave32. GEMM uses V_WMMA_F32_16X16X4_F32 (full f32 precision;
// the workload is memory-bound at ~1GB traffic vs 3.3 GFLOP, so f32 WMMA is free).

#define DFEAT 128
#define BN_EPS 1e-5f

typedef float v2f __attribute__((ext_vector_type(2)));
typedef float v8f __attribute__((ext_vector_type(8)));

// ---- 1: deg = 1 (self loop), zero BN accumulators (graph replay => must re-init) ----
__global__ void k_init(float* __restrict__ deg, float* __restrict__ sums,
                       float* __restrict__ sumsq, int N) {
    int i = blockIdx.x * blockDim.x + threadIdx.x;
    if (i < N) deg[i] = 1.0f;
    if (i < DFEAT) { sums[i] = 0.0f; sumsq[i] = 0.0f; }
}

// ---- 2: in-degree (dst side) ----
__global__ void k_degree(const long long* __restrict__ ei, float* __restrict__ deg, int E) {
    int e = blockIdx.x * blockDim.x + threadIdx.x;
    if (e < E) {
        long long dst = ei[(long long)E + e];
        atomicAdd(&deg[dst], 1.0f);
    }
}

// ---- 3: dinv = rsqrt(deg) in place (deg >= 1 always due to self loops) ----
__global__ void k_dinv(float* __restrict__ deg, int N) {
    int i = blockIdx.x * blockDim.x + threadIdx.x;
    if (i < N) deg[i] = rsqrtf(deg[i]);
}

// ---- 4: h = x @ W via V_WMMA_F32_16X16X4_F32 ----
// Block = 256 threads = 8 wave32s. Block computes a 16-row x 128-col stripe.
// A-tile (16x128) staged in LDS, shared by all 8 waves (8x reuse).
// Fragment packing per CDNA5 ISA 7.12.2:
//   A 16x4 f32 : lanes 0-15 row M=lane, v0=K0,v1=K1 ; lanes 16-31 v0=K2,v1=K3
//   B 4x16 f32 : lanes 0-15 col N=lane, v0=K0,v1=K1 ; lanes 16-31 v0=K2,v1=K3
//   C 16x16 f32: VGPR r -> M=r (lanes 0-15), M=r+8 (lanes 16-31), N=lane&15
__global__ __launch_bounds__(256) void k_gemm(const float* __restrict__ x,
                                              const float* __restrict__ W,
                                              float* __restrict__ h) {
    __shared__ float tileA[16 * DFEAT];  // 8 KB
    const int tid = threadIdx.x;
    const long long rowbase = (long long)blockIdx.x * 16;

    // Cooperative A-tile load: 512 float4, 2 per thread, fully coalesced.
    const float4* xs = (const float4*)(x + rowbase * DFEAT);
    float4* tv = (float4*)tileA;
    tv[tid]       = xs[tid];
    tv[tid + 256] = xs[tid + 256];
    __syncthreads();

    const int wave    = tid >> 5;
    const int lane    = tid & 31;
    const int half    = lane >> 4;   // 0: K{0,1}, 1: K{2,3}
    const int mn      = lane & 15;   // A row / B column within tile
    const int colbase = wave * 16;

    v8f c = {};
#pragma unroll
    for (int k = 0; k < DFEAT; k += 4) {
        // A fragment from LDS (ds_load_b64, 8B aligned: k + 2*half is even)
        v2f a = *(const v2f*)(&tileA[mn * DFEAT + k + 2 * half]);
        // B fragment from global W (lane-contiguous in columns -> coalesced per K-row)
        const float* wp = W + (size_t)(k + 2 * half) * DFEAT + colbase + mn;
        v2f bb;
        bb.x = wp[0];
        bb.y = wp[DFEAT];
        c = __builtin_amdgcn_wmma_f32_16x16x4_f32(
                /*neg_a=*/false, a, /*neg_b=*/false, bb,
                /*c_mod=*/(short)0, c, /*reuse_a=*/false, /*reuse_b=*/false);
    }
#pragma unroll
    for (int r = 0; r < 8; ++r) {
        h[(size_t)(rowbase + r + 8 * half) * DFEAT + colbase + mn] = c[r];
    }
}

// ---- 5: out = b + h * dinv^2   (self-loop term; out accumulates in d_out) ----
__global__ void k_selfinit(const float* __restrict__ h, const float* __restrict__ dinv,
                           const float* __restrict__ b, float* __restrict__ out,
                           long long total) {
    long long idx = (long long)blockIdx.x * blockDim.x + threadIdx.x;
    if (idx < total) {
        int i = (int)(idx >> 7);
        int d = (int)(idx & (DFEAT - 1));
        float di = dinv[i];
        out[idx] = b[d] + h[idx] * di * di;
    }
}

// ---- 6: scatter-add messages: out[dst] += h[src] * dinv[src]*dinv[dst] ----
// 128 threads per edge (one per feature), global_atomic_add_f32.
__global__ void k_scatter(const long long* __restrict__ ei,
                          const float* __restrict__ h,
                          const float* __restrict__ dinv,
                          float* __restrict__ out, int E) {
    long long t = (long long)blockIdx.x * blockDim.x + threadIdx.x;
    int e = (int)(t >> 7);
    int d = (int)(t & (DFEAT - 1));
    if (e < E) {
        long long s  = ei[e];
        long long dn = ei[(long long)E + e];
        float norm = dinv[s] * dinv[dn];
        atomicAdd(&out[(size_t)dn * DFEAT + d], h[(size_t)s * DFEAT + d] * norm);
    }
}

// ---- 7: per-feature sum / sum-of-squares (block partials + one atomic each) ----
__global__ __launch_bounds__(128) void k_stats(const float* __restrict__ out,
                                               float* __restrict__ sums,
                                               float* __restrict__ sumsq, int N) {
    int d = threadIdx.x;
    float s = 0.0f, q = 0.0f;
    for (int i = blockIdx.x; i < N; i += gridDim.x) {
        float v = out[(size_t)i * DFEAT + d];
        s += v;
        q += v * v;
    }
    atomicAdd(&sums[d], s);
    atomicAdd(&sumsq[d], q);
}

// ---- 8: BN (biased var) + ReLU + residual, in place on d_out ----
__global__ void k_final(float* __restrict__ out, const float* __restrict__ x,
                        const float* __restrict__ sums, const float* __restrict__ sumsq,
                        const float* __restrict__ gamma, const float* __restrict__ beta,
                        int N, long long total) {
    long long idx = (long long)blockIdx.x * blockDim.x + threadIdx.x;
    if (idx < total) {
        int d = (int)(idx & (DFEAT - 1));
        float invN = 1.0f / (float)N;
        float mean = sums[d] * invN;
        float var  = fmaxf(sumsq[d] * invN - mean * mean, 0.0f);
        float v = (out[idx] - mean) * rsqrtf(var + BN_EPS) * gamma[d] + beta[d];
        out[idx] = fmaxf(v, 0.0f) + x[idx];
    }
}

extern "C" void kernel_launch(void* const* d_in, const int* in_sizes, int n_in,
                              void* d_out, int out_size, void* d_ws, size_t ws_size,
                              hipStream_t stream) {
    const float*     x     = (const float*)d_in[0];
    const long long* ei    = (const long long*)d_in[1];  // int64 [2, E]
    const float*     W     = (const float*)d_in[2];
    const float*     b     = (const float*)d_in[3];
    const float*     gamma = (const float*)d_in[4];
    const float*     beta  = (const float*)d_in[5];
    float* out = (float*)d_out;

    const int N = in_sizes[0] / DFEAT;   // 100000 (multiple of 16)
    const int E = in_sizes[1] / 2;       // 600000
    const long long total = (long long)N * DFEAT;

    // Workspace layout: h[N*D] | dinv[N] | sums[D] | sumsq[D]  (~51.6 MB)
    float* h     = (float*)d_ws;
    float* dinv  = h + (size_t)N * DFEAT;
    float* sums  = dinv + N;
    float* sumsq = sums + DFEAT;

    const int B256 = 256;
    k_init    <<<(N + B256 - 1) / B256, B256, 0, stream>>>(dinv, sums, sumsq, N);
    k_degree  <<<(E + B256 - 1) / B256, B256, 0, stream>>>(ei, dinv, E);
    k_dinv    <<<(N + B256 - 1) / B256, B256, 0, stream>>>(dinv, N);
    k_gemm    <<<N / 16, B256, 0, stream>>>(x, W, h);
    k_selfinit<<<(int)((total + B256 - 1) / B256), B256, 0, stream>>>(h, dinv, b, out, total);
    {
        long long work = (long long)E * DFEAT;
        k_scatter<<<(int)((work + B256 - 1) / B256), B256, 0, stream>>>(ei, h, dinv, out, E);
    }
    k_stats   <<<512, 128, 0, stream>>>(out, sums, sumsq, N);
    k_final   <<<(int)((total + B256 - 1) / B256), B256, 0, stream>>>(out, x, sums, sumsq,
                                                                     gamma, beta, N, total);
}